// SelectiveSSM_19112604467498
// MI455X (gfx1250) — compile-verified
//
#include <hip/hip_runtime.h>
#include <stdint.h>

// Problem shape (fixed by the reference setup_inputs)
#define B_SZ   4
#define L_SZ   2048
#define D_SZ   1024
#define N_SZ   16
#define NC     16                 // number of time chunks
#define LC     (L_SZ / NC)        // 128 steps per chunk
#define DTILE  256                // d-channels (threads) per block
#define DTILES (D_SZ / DTILE)     // 4 d-tiles

// ---- CDNA5 async global->LDS helpers (gfx1250-specific path) ----
__device__ __forceinline__ uint32_t lds_u32(const void* p) {
    // Flat LDS aperture: low 32 bits of the flat address are the LDS byte offset.
    return (uint32_t)(uintptr_t)p;
}

// Stage 2048 floats (LC*N_SZ) from gsrc into lds, 8 floats per thread (2x b128).
// INST_OFFSET is added to BOTH the LDS and the global address per the ISA.
__device__ __forceinline__ void stage_chunk_async(const float* __restrict__ gsrc,
                                                  float* lds, int tid) {
    const float* g = gsrc + tid * 8;
    uint32_t l = lds_u32(&lds[tid * 8]);
    asm volatile("global_load_async_to_lds_b128 %0, %1, off"
                 :: "v"(l), "v"(g) : "memory");
    asm volatile("global_load_async_to_lds_b128 %0, %1, off offset:16"
                 :: "v"(l), "v"(g) : "memory");
}

__device__ __forceinline__ void wait_async_all() {
    asm volatile("s_wait_asynccnt 0" ::: "memory");
}

// ---------------- Phase A: per-chunk decay product P and zero-init state S ----
__global__ __launch_bounds__(DTILE) void ssm_phaseA(
    const float* __restrict__ x, const float* __restrict__ dt,
    const float* __restrict__ Bm, const float* __restrict__ A_log,
    float* __restrict__ Pws, float* __restrict__ Sws)
{
    __shared__ __align__(16) float ldsB[LC * N_SZ];

    const int tid  = threadIdx.x;
    const int bidx = blockIdx.x;
    const int tile = bidx % DTILES;
    const int c    = (bidx / DTILES) % NC;
    const int b    = bidx / (DTILES * NC);
    const int d    = tile * DTILE + tid;

    // async-stage B chunk for this (b, c): LC*N_SZ = 2048 floats
    stage_chunk_async(Bm + ((size_t)(b * L_SZ + c * LC)) * N_SZ, ldsB, tid);

    // A[d, n] = -exp(A_log[d, n])
    float An[N_SZ];
    {
        const float4* Ar = (const float4*)(A_log + (size_t)d * N_SZ);
        #pragma unroll
        for (int q = 0; q < 4; ++q) {
            float4 v = Ar[q];
            An[4*q+0] = -expf(v.x);
            An[4*q+1] = -expf(v.y);
            An[4*q+2] = -expf(v.z);
            An[4*q+3] = -expf(v.w);
        }
    }

    float st[N_SZ], Pp[N_SZ];
    #pragma unroll
    for (int n = 0; n < N_SZ; ++n) { st[n] = 0.0f; Pp[n] = 1.0f; }

    wait_async_all();
    __syncthreads();

    const size_t base = ((size_t)b * L_SZ + (size_t)c * LC) * D_SZ + d;
    const float* xp  = x  + base;
    const float* dtp = dt + base;

    // First touch of x/dt: default RT policy so lines land in the 192MB L2
    // and phase C's re-read hits L2 instead of HBM.
    #pragma unroll 4
    for (int t = 0; t < LC; ++t) {
        if (((t & 7) == 0) && (t + 16 < LC)) {
            __builtin_prefetch(xp  + (size_t)(t + 16) * D_SZ, 0, 0);
            __builtin_prefetch(dtp + (size_t)(t + 16) * D_SZ, 0, 0);
        }
        const float u   = xp [(size_t)t * D_SZ];
        const float dtv = dtp[(size_t)t * D_SZ];
        #pragma unroll
        for (int q = 0; q < 4; ++q) {
            const float4 bv = *(const float4*)&ldsB[t * N_SZ + 4 * q];
            float a;
            a = fmaf(An[4*q+0], dtv, 1.0f); Pp[4*q+0] *= a; st[4*q+0] = fmaf(a, st[4*q+0], u * bv.x);
            a = fmaf(An[4*q+1], dtv, 1.0f); Pp[4*q+1] *= a; st[4*q+1] = fmaf(a, st[4*q+1], u * bv.y);
            a = fmaf(An[4*q+2], dtv, 1.0f); Pp[4*q+2] *= a; st[4*q+2] = fmaf(a, st[4*q+2], u * bv.z);
            a = fmaf(An[4*q+3], dtv, 1.0f); Pp[4*q+3] *= a; st[4*q+3] = fmaf(a, st[4*q+3], u * bv.w);
        }
    }

    const size_t woff = (((size_t)(b * NC + c) * D_SZ) + d) * N_SZ;
    float4* Po = (float4*)(Pws + woff);
    float4* So = (float4*)(Sws + woff);
    #pragma unroll
    for (int q = 0; q < 4; ++q) {
        Po[q] = make_float4(Pp[4*q+0], Pp[4*q+1], Pp[4*q+2], Pp[4*q+3]);
        So[q] = make_float4(st[4*q+0], st[4*q+1], st[4*q+2], st[4*q+3]);
    }
}

// ------- Phase B: exclusive scan over chunks; S becomes each chunk's init state
__global__ __launch_bounds__(256) void ssm_phaseB(
    const float* __restrict__ Pws, float* __restrict__ Sws)
{
    const int i   = blockIdx.x * 256 + threadIdx.x;  // (b*D + d)*N + n
    const int b   = i / (D_SZ * N_SZ);
    const int rem = i % (D_SZ * N_SZ);
    float init = 0.0f;
    #pragma unroll
    for (int c = 0; c < NC; ++c) {
        const size_t off = (size_t)(b * NC + c) * (D_SZ * N_SZ) + rem;
        const float p = Pws[off];
        const float s = Sws[off];
        const float nxt = fmaf(p, init, s);
        Sws[off] = init;          // exclusive: initial state entering chunk c
        init = nxt;
    }
}

// ---------------- Phase C: replay each chunk from its exact init state, emit y
__global__ __launch_bounds__(DTILE) void ssm_phaseC(
    const float* __restrict__ x, const float* __restrict__ dt,
    const float* __restrict__ Bm, const float* __restrict__ Cm,
    const float* __restrict__ A_log, const float* __restrict__ Dv,
    const float* __restrict__ Sws, float* __restrict__ y)
{
    __shared__ __align__(16) float ldsB[LC * N_SZ];
    __shared__ __align__(16) float ldsC[LC * N_SZ];

    const int tid  = threadIdx.x;
    const int bidx = blockIdx.x;
    const int tile = bidx % DTILES;
    const int c    = (bidx / DTILES) % NC;
    const int b    = bidx / (DTILES * NC);
    const int d    = tile * DTILE + tid;

    stage_chunk_async(Bm + ((size_t)(b * L_SZ + c * LC)) * N_SZ, ldsB, tid);
    stage_chunk_async(Cm + ((size_t)(b * L_SZ + c * LC)) * N_SZ, ldsC, tid);

    float An[N_SZ];
    {
        const float4* Ar = (const float4*)(A_log + (size_t)d * N_SZ);
        #pragma unroll
        for (int q = 0; q < 4; ++q) {
            float4 v = Ar[q];
            An[4*q+0] = -expf(v.x);
            An[4*q+1] = -expf(v.y);
            An[4*q+2] = -expf(v.z);
            An[4*q+3] = -expf(v.w);
        }
    }
    const float Dd = Dv[d];

    // initial state for this chunk (written by phase B)
    float st[N_SZ];
    {
        const float4* Si = (const float4*)(Sws + (((size_t)(b * NC + c) * D_SZ) + d) * N_SZ);
        #pragma unroll
        for (int q = 0; q < 4; ++q) {
            float4 v = Si[q];
            st[4*q+0] = v.x; st[4*q+1] = v.y; st[4*q+2] = v.z; st[4*q+3] = v.w;
        }
    }

    wait_async_all();
    __syncthreads();

    const size_t base = ((size_t)b * L_SZ + (size_t)c * LC) * D_SZ + d;
    const float* xp  = x  + base;
    const float* dtp = dt + base;
    float*       yp  = y  + base;

    // Last touch of x/dt and streaming y: non-temporal so we don't evict
    // live L2 lines; y is write-once streamed out.
    #pragma unroll 4
    for (int t = 0; t < LC; ++t) {
        const float u   = __builtin_nontemporal_load(xp  + (size_t)t * D_SZ);
        const float dtv = __builtin_nontemporal_load(dtp + (size_t)t * D_SZ);
        float acc0 = Dd * u, acc1 = 0.0f, acc2 = 0.0f, acc3 = 0.0f;
        #pragma unroll
        for (int q = 0; q < 4; ++q) {
            const float4 bv = *(const float4*)&ldsB[t * N_SZ + 4 * q];
            const float4 cv = *(const float4*)&ldsC[t * N_SZ + 4 * q];
            float a;
            a = fmaf(An[4*q+0], dtv, 1.0f); st[4*q+0] = fmaf(a, st[4*q+0], u * bv.x); acc0 = fmaf(cv.x, st[4*q+0], acc0);
            a = fmaf(An[4*q+1], dtv, 1.0f); st[4*q+1] = fmaf(a, st[4*q+1], u * bv.y); acc1 = fmaf(cv.y, st[4*q+1], acc1);
            a = fmaf(An[4*q+2], dtv, 1.0f); st[4*q+2] = fmaf(a, st[4*q+2], u * bv.z); acc2 = fmaf(cv.z, st[4*q+2], acc2);
            a = fmaf(An[4*q+3], dtv, 1.0f); st[4*q+3] = fmaf(a, st[4*q+3], u * bv.w); acc3 = fmaf(cv.w, st[4*q+3], acc3);
        }
        const float acc = (acc0 + acc1) + (acc2 + acc3);
        __builtin_nontemporal_store(acc, yp + (size_t)t * D_SZ);
    }
}

extern "C" void kernel_launch(void* const* d_in, const int* in_sizes, int n_in,
                              void* d_out, int out_size, void* d_ws, size_t ws_size,
                              hipStream_t stream) {
    const float* x    = (const float*)d_in[0];
    const float* dt   = (const float*)d_in[1];
    const float* Bm   = (const float*)d_in[2];
    const float* Cm   = (const float*)d_in[3];
    const float* Alog = (const float*)d_in[4];
    const float* Dv   = (const float*)d_in[5];
    float* y = (float*)d_out;

    // Workspace: P then S, each B*NC*D*N floats (4 MB each; 8 MB total)
    float* Pws = (float*)d_ws;
    float* Sws = Pws + (size_t)B_SZ * NC * D_SZ * N_SZ;

    const dim3 grid(B_SZ * NC * DTILES);   // 256 blocks
    const dim3 blk(DTILE);                 // 256 threads (8 wave32)

    ssm_phaseA<<<grid, blk, 0, stream>>>(x, dt, Bm, Alog, Pws, Sws);
    ssm_phaseB<<<dim3((B_SZ * D_SZ * N_SZ) / 256), dim3(256), 0, stream>>>(Pws, Sws);
    ssm_phaseC<<<grid, blk, 0, stream>>>(x, dt, Bm, Cm, Alog, Dv, Sws, y);
}